// CustomLossNN_52664888984291
// MI455X (gfx1250) — compile-verified
//
#include <hip/hip_runtime.h>
#include <stdint.h>

// ---------------------------------------------------------------------------
// CustomLossNN: cross-entropy over [B=4096, C=32000] f32 logits + constant
// target-uniqueness penalty (shape-determined: 10*B*(C-1)).
// TDM double-buffered: two half-row tensor_load_to_lds per workgroup
// (64KB each), max-pass overlaps the second half's DMA. Exact two-pass
// logsumexp out of LDS with ds_load_b128 reads. HBM-bound: 524MB @ 23.3TB/s.
// ---------------------------------------------------------------------------

typedef unsigned int v4u __attribute__((ext_vector_type(4)));
typedef int          v8i __attribute__((ext_vector_type(8)));
typedef int          v4i __attribute__((ext_vector_type(4)));

#define BLOCK 256
#define NWAVES (BLOCK / 32)

// Issue one 1-D TDM descriptor: n f32 elements, global ga -> LDS lds_off.
// th: cache policy (1 = TH_LOAD_NT; stream-once data must not thrash L2).
__device__ __forceinline__ void tdm_load_1d(uint64_t ga, uint32_t lds_off, int n) {
  const uint32_t n16lo = (uint32_t)n & 0xFFFFu;
  const uint32_t n16hi = ((uint32_t)n >> 16) & 0xFFFFu;
  v4u g0;
  g0.x = 1u;                                        // count=1, user descriptor
  g0.y = lds_off;                                   // lds_addr (bytes, 16B aligned)
  g0.z = (uint32_t)ga;                              // global_addr[31:0]
  g0.w = ((uint32_t)(ga >> 32) & 0x01FFFFFFu)       // global_addr[56:32]
         | (2u << 30);                              // type = 2 ("image")
  v8i g1;
  g1[0] = (int)(2u << 16);                          // data_size = 2 (4 bytes)
  g1[1] = (int)(n16lo << 16);                       // tensor_dim0[15:0]
  g1[2] = (int)(n16hi | (1u << 16));                // tensor_dim0[31:16], tensor_dim1=1
  g1[3] = (int)(n16lo << 16);                       // tile_dim0 = n
  g1[4] = 1;                                        // tile_dim1 = 1
  g1[5] = n;                                        // tensor_dim0_stride
  g1[6] = 0;
  g1[7] = 0;
  v4i g2 = {0, 0, 0, 0};
  v4i g3 = {0, 0, 0, 0};
  v8i g4 = {0, 0, 0, 0, 0, 0, 0, 0};
  __builtin_amdgcn_tensor_load_to_lds(g0, g1, g2, g3, g4, /*cpol: TH_NT*/ 1);
}

__global__ void ce_row_kernel(const float* __restrict__ logits,
                              const int*   __restrict__ targets,
                              float*       __restrict__ row_loss,
                              int C) {
  extern __shared__ __align__(16) float smem[];   // C floats (128 KB), 16B aligned
  __shared__ float red[NWAVES];
  __shared__ float bcast;

  const int row  = blockIdx.x;
  const int tid  = threadIdx.x;
  const int half = C >> 1;                        // C = 32000 -> 16000 (fits tile_dim0)

  // ---- TDM: two pipelined half-row DMAs, global -> LDS ----
  if (tid < 32) {
    const uint64_t ga      = (uint64_t)(uintptr_t)(logits + (size_t)row * (size_t)C);
    const uint32_t lds_off = (uint32_t)(uintptr_t)smem;
    tdm_load_1d(ga,                          lds_off,                          half);
    tdm_load_1d(ga + (uint64_t)half * 4u,    lds_off + (uint32_t)half * 4u,    C - half);
    __builtin_amdgcn_s_wait_tensorcnt(1);         // first half resident
  }
  __syncthreads();                                // half 0 visible to all waves

  // ---- Pass 1a: max over half 0 (overlaps half-1 DMA) ----
  const float4* smem4 = (const float4*)smem;
  const int nv0 = half >> 2;
  float m = -3.402823466e38f;
  for (int i = tid; i < nv0; i += BLOCK) {
    float4 v = smem4[i];
    m = fmaxf(m, fmaxf(fmaxf(v.x, v.y), fmaxf(v.z, v.w)));
  }

  if (tid < 32) __builtin_amdgcn_s_wait_tensorcnt(0);  // second half resident
  __syncthreads();                                // half 1 visible to all waves

  // ---- Pass 1b: max over half 1 ----
  const int nv = C >> 2;
  for (int i = nv0 + tid; i < nv; i += BLOCK) {
    float4 v = smem4[i];
    m = fmaxf(m, fmaxf(fmaxf(v.x, v.y), fmaxf(v.z, v.w)));
  }
  for (int i = (nv << 2) + tid; i < C; i += BLOCK) m = fmaxf(m, smem[i]);  // tail

  #pragma unroll
  for (int off = 16; off > 0; off >>= 1) m = fmaxf(m, __shfl_xor(m, off, 32));
  if ((tid & 31) == 0) red[tid >> 5] = m;
  __syncthreads();
  if (tid == 0) {
    float v = red[0];
    #pragma unroll
    for (int w = 1; w < NWAVES; ++w) v = fmaxf(v, red[w]);
    bcast = v;
  }
  __syncthreads();
  const float rowmax = bcast;
  __syncthreads();                                // protect red[] before reuse

  // ---- Pass 2: sum of exp(x - max) ----
  float s = 0.0f;
  for (int i = tid; i < nv; i += BLOCK) {
    float4 v = smem4[i];
    s += __expf(v.x - rowmax);
    s += __expf(v.y - rowmax);
    s += __expf(v.z - rowmax);
    s += __expf(v.w - rowmax);
  }
  for (int i = (nv << 2) + tid; i < C; i += BLOCK) s += __expf(smem[i] - rowmax);

  #pragma unroll
  for (int off = 16; off > 0; off >>= 1) s += __shfl_xor(s, off, 32);
  if ((tid & 31) == 0) red[tid >> 5] = s;
  __syncthreads();
  if (tid == 0) {
    float tot = 0.0f;
    #pragma unroll
    for (int w = 0; w < NWAVES; ++w) tot += red[w];
    const int t = targets[row];
    row_loss[row] = rowmax + __logf(tot) - smem[t];
  }
}

// Deterministic fixed-order reduction of per-row losses; folds in the
// host-computed constant penalty.
__global__ void finalize_kernel(const float* __restrict__ row_loss,
                                float* __restrict__ out,
                                int B, float penalty) {
  __shared__ float red[256];
  const int tid = threadIdx.x;
  float s = 0.0f;
  for (int i = tid; i < B; i += 256) s += row_loss[i];   // fixed order per lane
  red[tid] = s;
  __syncthreads();
  #pragma unroll
  for (int off = 128; off > 0; off >>= 1) {
    if (tid < off) red[tid] += red[tid + off];
    __syncthreads();
  }
  if (tid == 0) out[0] = red[0] / (float)B + penalty;
}

extern "C" void kernel_launch(void* const* d_in, const int* in_sizes, int n_in,
                              void* d_out, int out_size, void* d_ws, size_t ws_size,
                              hipStream_t stream) {
  const float* logits  = (const float*)d_in[0];
  const int*   targets = (const int*)d_in[1];
  const int B = in_sizes[1];                 // 4096
  const int C = in_sizes[0] / B;             // 32000

  float* ws  = (float*)d_ws;                 // B row losses (16 KB)
  float* out = (float*)d_out;

  // Penalty is shape-constant: uniq==1 per row -> repeated = C-1 each row.
  const double pen = 10.0 * (double)B * (double)(C - 1);

  ce_row_kernel<<<B, BLOCK, (size_t)C * sizeof(float), stream>>>(logits, targets, ws, C);
  finalize_kernel<<<1, 256, 0, stream>>>(ws, out, B, (float)pen);
}